// HungarianMatcher_56092272886326
// MI455X (gfx1250) — compile-verified
//
#include <hip/hip_runtime.h>
#include <hip/hip_bf16.h>

typedef float v2f __attribute__((ext_vector_type(2)));
typedef float v8f __attribute__((ext_vector_type(8)));

#define NN 512
#define NP 513
#define NWAVES (NN / 32)
#define INFV 1.0e9f

// ---------------------------------------------------------------------------
// Kernel 1: cost = sim + iou via WMMA (bitwise exact: identity-A trick).
// One wave per 16x16 tile. D = sum_kk I[:,4kk:4kk+4] x sim[4kk:4kk+4, :] + iou
// Each output element gets exactly one 1.0*sim product added to iou -> exact.
// ---------------------------------------------------------------------------
__global__ void hm_cost_wmma(const float* __restrict__ sim,
                             const float* __restrict__ iou,
                             float* __restrict__ cost) {
  const int wave = threadIdx.x >> 5;
  const int lane = threadIdx.x & 31;
  const int tile = blockIdx.x * (blockDim.x >> 5) + wave;  // 0..1023
  const int tr = tile >> 5;        // tile row (32 tiles)
  const int tc = tile & 31;        // tile col
  const int r0 = tr << 4;
  const int c0 = tc << 4;
  const int half = lane >> 4;      // 0: lanes 0-15, 1: lanes 16-31
  const int l15 = lane & 15;

  // C/D layout: vgpr v -> row M = v + 8*half, col N = l15
  v8f c;
#pragma unroll
  for (int v = 0; v < 8; ++v)
    c[v] = iou[(size_t)(r0 + v + half * 8) * NN + (c0 + l15)];

#pragma unroll
  for (int kk = 0; kk < 4; ++kk) {
    // A (16x4 f32, documented): lane m=l15; vgpr0 K = 2*half, vgpr1 K = 2*half+1
    v2f a;
    a.x = (l15 == kk * 4 + half * 2 + 0) ? 1.0f : 0.0f;
    a.y = (l15 == kk * 4 + half * 2 + 1) ? 1.0f : 0.0f;
    // B (4x16 f32): vgpr v -> K = v + 2*half, N = l15 (mirrors C/D half-wave split)
    v2f b;
    b.x = sim[(size_t)(r0 + kk * 4 + half * 2 + 0) * NN + (c0 + l15)];
    b.y = sim[(size_t)(r0 + kk * 4 + half * 2 + 1) * NN + (c0 + l15)];
    c = __builtin_amdgcn_wmma_f32_16x16x4_f32(false, a, false, b,
                                              (short)0, c, false, false);
  }

#pragma unroll
  for (int v = 0; v < 8; ++v)
    cost[(size_t)(r0 + v + half * 8) * NN + (c0 + l15)] = c[v];
}

// first-index-tie-break min combine (matches jnp.argmin semantics)
__device__ __forceinline__ void minred(float& bv, int& bj, float ov, int oj) {
  if (ov < bv || (ov == bv && oj < bj)) { bv = ov; bj = oj; }
}

// ---------------------------------------------------------------------------
// Kernel 2: exact Hungarian (JV shortest augmenting path) on -cost.
// Single workgroup of 512 threads (16 wave32s); solver state in LDS.
// Thread t owns column j = t+1. Argmin: wave32 shuffle reduce + one
// cross-wave stage -> only 3 barriers per inner step.
// ---------------------------------------------------------------------------
__global__ void __launch_bounds__(NN)
hm_solver(const float* __restrict__ cost, float* __restrict__ out) {
  __shared__ float u[NP];
  __shared__ float v[NP];
  __shared__ float minv[NP];
  __shared__ int   p[NP];
  __shared__ int   way[NP];
  __shared__ int   used[NP];
  __shared__ float wval[NWAVES];
  __shared__ int   widx[NWAVES];

  const int t    = threadIdx.x;
  const int j    = t + 1;
  const int lane = t & 31;
  const int wid  = t >> 5;

  // init duals and matching
  u[t] = 0.0f; v[t] = 0.0f; p[t] = 0;
  if (t == 0) { u[NN] = 0.0f; v[NN] = 0.0f; p[NN] = 0; }
  __syncthreads();

  for (int i = 1; i <= NN; ++i) {
    // per-row reset; col 0 is pre-marked used: body always runs at j0=0
    minv[j] = INFV; used[j] = 0; way[j] = 0;
    if (t == 0) { p[0] = i; minv[0] = INFV; used[0] = 1; way[0] = 0; }
    __syncthreads();

    int j0 = 0;
    for (;;) {
      if (p[j0] == 0) break;          // uniform: free column reached

      const int   i0  = p[j0];
      const float ui0 = u[i0];

      // column scan (thread-local minv/way/v; used[j0] marked last step)
      float cand;
      if (!used[j]) {
        float cur = (-cost[(size_t)(i0 - 1) * NN + (j - 1)] - ui0) - v[j];
        if (cur < minv[j]) { minv[j] = cur; way[j] = j0; }
        cand = minv[j];
      } else {
        cand = INFV;
      }

      // stage 1: wave32 shuffle argmin (no barriers)
      float bv = cand;
      int   bj = j;
#pragma unroll
      for (int off = 16; off > 0; off >>= 1) {
        float ov = __shfl_xor(bv, off, 32);
        int   oj = __shfl_xor(bj, off, 32);
        minred(bv, bj, ov, oj);
      }
      if (lane == 0) { wval[wid] = bv; widx[wid] = bj; }
      __syncthreads();                                   // barrier 1

      // stage 2: wave 0 reduces the 16 per-wave partials
      if (wid == 0) {
        float v2 = (lane < NWAVES) ? wval[lane] : INFV;
        int   j2 = (lane < NWAVES) ? widx[lane] : 0x7fffffff;
#pragma unroll
        for (int off = NWAVES >> 1; off > 0; off >>= 1) {
          float ov = __shfl_xor(v2, off, 32);
          int   oj = __shfl_xor(j2, off, 32);
          minred(v2, j2, ov, oj);
        }
        if (lane == 0) { wval[0] = v2; widx[0] = j2; }
      }
      __syncthreads();                                   // barrier 2

      const float delta = wval[0];
      const int   j1    = widx[0];
      const int   nr    = p[j1];

      // prefetch the row we will scan next step
      if (nr != 0)
        __builtin_prefetch(cost + (size_t)(nr - 1) * NN + t, 0, 3);

      // dual updates (conflict-free scatter: matched rows are distinct)
      if (used[j]) { u[p[j]] += delta; v[j] -= delta; }
      else         { minv[j] -= delta; }
      if (t == 0) {
        u[p[0]] += delta; v[0] -= delta;   // col 0 is always used in-loop
        if (nr != 0) used[j1] = 1;         // pre-mark next step's j0
      }
      __syncthreads();                                   // barrier 3

      j0 = j1;
    }

    // augment along the alternating path
    if (t == 0) {
      int jj = j0;
      while (jj != 0) {
        int jp = way[jj];
        p[jj] = p[jp];
        jj = jp;
      }
    }
    __syncthreads();
  }

  // outputs: row_ind = arange, col_ind[p[j]-1] = j-1 (written as floats)
  float* row_out = out + (size_t)NN * NN;
  float* col_out = row_out + NN;
  row_out[t] = (float)t;
  col_out[p[j] - 1] = (float)(j - 1);
}

// ---------------------------------------------------------------------------
extern "C" void kernel_launch(void* const* d_in, const int* in_sizes, int n_in,
                              void* d_out, int out_size, void* d_ws, size_t ws_size,
                              hipStream_t stream) {
  (void)in_sizes; (void)n_in; (void)d_ws; (void)ws_size; (void)out_size;
  const float* sim = (const float*)d_in[0];
  const float* iou = (const float*)d_in[1];
  float* out = (float*)d_out;

  // 1024 tiles of 16x16, 8 waves (256 threads) per block -> 128 blocks
  hm_cost_wmma<<<dim3(128), dim3(256), 0, stream>>>(sim, iou, out);
  // single-workgroup sequential-augmenting-path solver
  hm_solver<<<dim3(1), dim3(NN), 0, stream>>>(out, out);
}